// SNNEventModel_86260123173248
// MI455X (gfx1250) — compile-verified
//
#include <hip/hip_runtime.h>
#include <hip/hip_bf16.h>

// ---------------------------------------------------------------------------
// SNN event model, fully fused: causal conv1d (1->128ch, K=16) + bias computed
// per 32-wide time chunk via V_WMMA_F32_16X16X4_F32 directly into LDS, then
// the sequential LIF scan + logits on the same block. One block = one batch.
// Outputs (f32, concat flat): I, v_pre, v, s [64,128,8192] each; logits [64,8192].
// Traffic: read x (2 MB) + write 4*256 MB + 2 MB  ~= 1.08 GB -> HBM-floor bound.
// Chunk 0 is peeled so the steady-state inner loop has no pad guards (plain
// loads + 16 WMMAs), avoiding the exec-mask branches clang emits for
// conditional loads.
// ---------------------------------------------------------------------------

typedef __attribute__((ext_vector_type(2))) float v2f;
typedef __attribute__((ext_vector_type(8))) float v8f;

#define BATCH   64
#define NF      128
#define LSEQ    8192
#define KTAPS   16
#define LIF_K   0.2f      // DT / TAU
#define THR     1.0f

#define CHUNK 32
#define LROW  33          // CHUNK + 1 pad -> conflict-free column reads

// Conv for one 128f x 32t chunk: 2 filter-tiles x 2 time-tiles per wave,
// 16 v_wmma_f32_16x16x4_f32 total per wave. GUARD only needed for t0 == 0.
template <bool GUARD>
__device__ __forceinline__ void conv_chunk(const float* __restrict__ xb,
                                           int t0, const v2f aw[2][4],
                                           const v8f cb[2],
                                           int wave, int hi, int n,
                                           float* __restrict__ in_t,
                                           float* __restrict__ I_out,
                                           size_t rowbase)
{
#pragma unroll
    for (int ft = 0; ft < 2; ++ft) {
        const int f0 = (wave << 5) + (ft << 4);
#pragma unroll
        for (int tt = 0; tt < 2; ++tt) {
            const int tbase = t0 + (tt << 4);
            v8f c = cb[ft];
#pragma unroll
            for (int kc = 0; kc < 4; ++kc) {
                const int k0 = (kc << 2) + (hi << 1);
                // B[k][t] = x[b, tbase + n + k - (K-1)]; max index <= t0+31,
                // so only the left (causal-pad) edge can be OOB.
                const int ti0 = tbase + n + k0 - (KTAPS - 1);
                v2f bf;
                if (GUARD) {
                    // Unconditional load from clamped address + value select:
                    // lowers to v_max/v_cndmask, NOT exec-masked branches.
                    int c0 = ti0 < 0 ? 0 : ti0;
                    int c1 = ti0 < -1 ? 0 : ti0 + 1;
                    float x0 = xb[c0];
                    float x1 = xb[c1];
                    bf[0] = (ti0 >= 0)  ? x0 : 0.0f;
                    bf[1] = (ti0 >= -1) ? x1 : 0.0f;
                } else {
                    bf[0] = xb[ti0];
                    bf[1] = xb[ti0 + 1];
                }
                c = __builtin_amdgcn_wmma_f32_16x16x4_f32(
                        /*neg_a=*/false, aw[ft][kc], /*neg_b=*/false, bf,
                        /*c_mod=*/(short)0, c,
                        /*reuse_a=*/false, /*reuse_b=*/false);
            }
            // D: f = f0 + v + 8*hi, t = tbase + n. Write to LDS staging tile
            // and stream to global I (16 consecutive floats per half-wave).
#pragma unroll
            for (int v = 0; v < 8; ++v) {
                const int f = f0 + v + (hi << 3);
                const float d = c[v];
                in_t[f * LROW + (tt << 4) + n] = d;
                I_out[(rowbase + f) * LSEQ + tbase + n] = d;
            }
        }
    }
}

__global__ void snn_fused(const float* __restrict__ x,
                          const float* __restrict__ W,
                          const float* __restrict__ bias,
                          float* __restrict__ I_out,
                          float* __restrict__ vpre_out,
                          float* __restrict__ v_out,
                          float* __restrict__ s_out,
                          float* __restrict__ logits)
{
    extern __shared__ float smem[];
    float* in_t = smem;                 // [128][LROW] : I chunk (from WMMA)
    float* vp_t = smem + NF * LROW;     // [128][LROW] : v_pre chunk

    const int tid  = threadIdx.x;       // 0..127 : filter row for the scan
    const int lane = tid & 31;
    const int wave = tid >> 5;          // 0..3 : owns filters [32w, 32w+32)
    const int b    = blockIdx.x;        // 0..63 : batch
    const size_t rowbase = (size_t)b * NF;

    const int hi = lane >> 4;           // K-half select
    const int n  = lane & 15;           // A: M row (filter) ; B: N col (time)

    const float* xb = x + (size_t)b * LSEQ;

    // ---- Preload W fragments (A: 16 filters x 4 taps per k-chunk) and
    // ---- bias-seeded C fragments; constant across all 256 chunks.
    v2f aw[2][4];
    v8f cb[2];
#pragma unroll
    for (int ft = 0; ft < 2; ++ft) {
        const int f0 = (wave << 5) + (ft << 4);
#pragma unroll
        for (int kc = 0; kc < 4; ++kc) {
            const int k0 = (kc << 2) + (hi << 1);
            aw[ft][kc][0] = W[(f0 + n) * KTAPS + k0];
            aw[ft][kc][1] = W[(f0 + n) * KTAPS + k0 + 1];
        }
#pragma unroll
        for (int v = 0; v < 8; ++v)
            cb[ft][v] = bias[f0 + v + (hi << 3)];
    }

    float vmem = 0.0f;   // carried membrane potential for my filter row

    for (int t0 = 0; t0 < LSEQ; t0 += CHUNK) {
        // ---- Conv chunk via WMMA (EXEC all-ones: full block, no divergence).
        if (t0 == 0)      // uniform scalar branch; peel keeps main loop clean
            conv_chunk<true >(xb, t0, aw, cb, wave, hi, n, in_t, I_out, rowbase);
        else
            conv_chunk<false>(xb, t0, aw, cb, wave, hi, n, in_t, I_out, rowbase);
        __syncthreads();

        // ---- Sequential LIF over this chunk for my row (stride-33 LDS
        // column reads are conflict-free across the wave).
        float vloc = vmem;
#pragma unroll
        for (int j = 0; j < CHUNK; ++j) {
            float it   = in_t[tid * LROW + j];
            float vpre = fmaf(LIF_K, it - vloc, vloc);
            vloc = (vpre >= THR) ? 0.0f : vpre;   // hard reset
            vp_t[tid * LROW + j] = vpre;
        }
        vmem = vloc;
        __syncthreads();

        // ---- Coalesced streaming stores of v_pre / v / s (v and s
        // recomputed from v_pre; saves two LDS tiles).
#pragma unroll 4
        for (int i = 0; i < 32; ++i) {
            int e = tid + NF * i;           // 0..4095
            int r = e >> 5;
            int j = e & (CHUNK - 1);
            float vpre = vp_t[r * LROW + j];
            float fire = (vpre >= THR) ? 1.0f : 0.0f;
            size_t g = (rowbase + r) * LSEQ + t0 + j;
            vpre_out[g] = vpre;
            v_out[g]    = vpre * (1.0f - fire);
            s_out[g]    = fire;
        }

        // ---- logits[b, t0+j] = max over 128 filters of v_pre - THR
        if (tid < CHUNK) {
            int j = tid;
            float m = -3.402823466e+38f;
#pragma unroll 8
            for (int r = 0; r < NF; ++r)
                m = fmaxf(m, vp_t[r * LROW + j]);
            logits[(size_t)b * LSEQ + t0 + j] = m - THR;
        }
        __syncthreads();   // tiles reused next chunk
    }
}

// ---------------------------------------------------------------------------
extern "C" void kernel_launch(void* const* d_in, const int* in_sizes, int n_in,
                              void* d_out, int out_size, void* d_ws, size_t ws_size,
                              hipStream_t stream)
{
    const float* x    = (const float*)d_in[0];   // [64,1,8192]
    const float* W    = (const float*)d_in[1];   // [128,1,16]
    const float* bias = (const float*)d_in[2];   // [128]

    float* out = (float*)d_out;
    const size_t BFL = (size_t)BATCH * NF * LSEQ;
    float* I      = out;
    float* vpre   = out + BFL;
    float* vv     = out + 2 * BFL;
    float* ss     = out + 3 * BFL;
    float* logits = out + 4 * BFL;

    // One block per batch; 4 waves cover the 128 filters; ~33 KB LDS.
    size_t shmem = 2ull * NF * LROW * sizeof(float);
    snn_fused<<<BATCH, NF, shmem, stream>>>(x, W, bias, I, vpre, vv, ss, logits);
}